// DWT_3D_60327110639923
// MI455X (gfx1250) — compile-verified
//
#include <hip/hip_runtime.h>
#include <stdint.h>

// Fused 2-level 3D Haar DWT + LIF spiking pipeline for MI455X (gfx1250).
//
// x: (2,32,64,128,128) f32.  Outputs concatenated in d_out:
//   out0  = sub2[0] : (2,32,16,32,32)      @ 0          (1,048,576 elems)
//   high1           : (2,7,32,32,64,64)    @ 1,048,576  (58,720,256 elems)
//   high2           : (2,7,32,16,32,32)    @ 59,768,832 (7,340,032 elems)
//
// Memory-bound (512 MB mandatory traffic, ~2 GFLOP -> ~4 flop/byte). One
// thread owns a 4x4 (h,w) input patch, so both DWT levels and all 13 LIF
// depth-scans are thread-local (membrane potentials in VGPRs). The input is
// streamed over depth through an 8-deep LDS ring filled with
// global_load_async_to_lds_b128 (ASYNCcnt): 512 blocks x 7 tiles x 8 KB
// = 28 MB of DMA in flight to cover HBM latency at 23.3 TB/s with only
// 2048 wave32s. One barrier per depth step (issue-after-compute makes the
// trailing barrier redundant). Outputs leave via nontemporal stores.

typedef float v2f __attribute__((ext_vector_type(2)));
typedef float v4f __attribute__((ext_vector_type(4)));

#if __has_builtin(__builtin_amdgcn_s_wait_asynccnt)
#define WAITA(n) __builtin_amdgcn_s_wait_asynccnt(n)
#else
#define WAITA(n) asm volatile("s_wait_asynccnt " #n ::: "memory")
#endif

__device__ __forceinline__ float lif_step(float& v, float x) {
  // v' = v + (x - v)/1.75 ; spike = v' >= 0.35 ; hard reset to 0.
  float vn = v + (x - v) * 0.5714285714285714f;
  float s  = (vn >= 0.35f) ? 1.0f : 0.0f;
  v        = (vn >= 0.35f) ? 0.0f : vn;
  return s;
}

__global__ void __launch_bounds__(128)
dwt3d_lif_fused(const float* __restrict__ x, float* __restrict__ out) {
  extern __shared__ __align__(16) float smem[];   // 8 ring buffers x 2048 f32 = 64 KB

  const uint32_t t     = threadIdx.x;
  const uint32_t tx    = t & 31u;                 // q lane: final col [0,32)
  const uint32_t ty    = t >> 5;                  // wave id: 4 rows of the strip
  const uint32_t bid   = blockIdx.x;
  const uint32_t slice = bid >> 3;                // n*32 + c
  const uint32_t si    = bid & 7u;                // 4-row strip of final P
  const uint32_t n     = slice >> 5, ch = slice & 31u;
  const uint32_t Pg    = si * 4u + ty;            // final-level row [0,32)
  const uint32_t Qg    = tx;                      // final-level col [0,32)

  // ---- async global->LDS staging: per d a 16x128 f32 tile (8 KB) ----
  const uint32_t f0    = t * 16u;                 // this thread's 16B slot
  const uint32_t lbase = (uint32_t)(uintptr_t)smem;
  uint64_t g[4];
  {
    const uint64_t gslice = (uint64_t)(uintptr_t)x +
        4ull * ((uint64_t)n * 33554432ull + (uint64_t)ch * 1048576ull +
                (uint64_t)(si * 16u) * 128ull);
    #pragma unroll
    for (int k = 0; k < 4; ++k) {
      uint32_t fk = f0 + (uint32_t)k * 2048u;            // byte in tile
      g[k] = gslice + (uint64_t)(fk >> 9) * 512ull       // row * (W*4B)
                    + (uint64_t)(fk & 511u);             // col bytes
    }
  }

  auto issue = [&](uint32_t buf, uint64_t dbyte) {
    const uint32_t bb = lbase + buf * 8192u + f0;
    #pragma unroll
    for (int k = 0; k < 4; ++k) {
      uint32_t la = bb + (uint32_t)k * 2048u;
      uint64_t ga = g[k] + dbyte;
      asm volatile("global_load_async_to_lds_b128 %0, %1, off"
                   :: "v"(la), "v"(ga) : "memory");
    }
  };

  // ---- persistent LIF membrane potentials (all in VGPRs) ----
  float vL[2][4]    = {};   // stage-1a low  (p2, w)
  float vH[2][4]    = {};   // stage-1a high (p2, w)
  float vS[4][2][2] = {};   // stage-1b LL,LH,HL,HH at (p2,q2)
  float ev1[4][2][2];       // even-d spike stash for depth pairing
  float vLLL[2][2]  = {};   // LIF on sub1[0]
  float vL2[2] = {}, vH2[2] = {};   // stage-2a
  float vS2[4] = {};        // stage-2b LL2..HH2
  float ev2[4];             // even-e stash

  // ---- output bases ----
  float* h1b = out + 1048576u
             + (size_t)n * 29360128u + (size_t)ch * 131072u
             + (size_t)(2u * Pg) * 64u + (size_t)(2u * Qg);
  float* o0b = out + (size_t)n * 524288u + (size_t)ch * 16384u
             + (size_t)Pg * 32u + (size_t)Qg;
  float* h2b = out + 59768832u
             + (size_t)n * 3670016u + (size_t)ch * 16384u
             + (size_t)Pg * 32u + (size_t)Qg;

  const float RC = 0.70710678118654752f;

  auto stage1 = [&](const float* tp, float (&s1)[4][2][2]) {
    float xv[4][4];
    #pragma unroll
    for (int r = 0; r < 4; ++r) {
      v4f rv = *(const v4f*)(tp + (4u * ty + (uint32_t)r) * 128u + 4u * tx);
      xv[r][0] = rv[0]; xv[r][1] = rv[1]; xv[r][2] = rv[2]; xv[r][3] = rv[3];
    }
    float sL[2][4], sH[2][4];
    #pragma unroll
    for (int p2 = 0; p2 < 2; ++p2)
      #pragma unroll
      for (int w = 0; w < 4; ++w) {
        float a = xv[2 * p2][w], b = xv[2 * p2 + 1][w];
        sL[p2][w] = lif_step(vL[p2][w], (a + b) * RC);
        sH[p2][w] = lif_step(vH[p2][w], (a - b) * RC);
      }
    #pragma unroll
    for (int p2 = 0; p2 < 2; ++p2)
      #pragma unroll
      for (int q2 = 0; q2 < 2; ++q2) {
        float la = sL[p2][2 * q2], lb = sL[p2][2 * q2 + 1];
        float ha = sH[p2][2 * q2], hb = sH[p2][2 * q2 + 1];
        s1[0][p2][q2] = lif_step(vS[0][p2][q2], (la + lb) * RC);
        s1[1][p2][q2] = lif_step(vS[1][p2][q2], (la - lb) * RC);
        s1[2][p2][q2] = lif_step(vS[2][p2][q2], (ha + hb) * RC);
        s1[3][p2][q2] = lif_step(vS[3][p2][q2], (ha - hb) * RC);
      }
  };

  auto evenStep = [&](uint32_t buf) {
    stage1((const float*)(smem + (size_t)buf * 2048u), ev1);
  };

  auto oddStep = [&](uint32_t buf, int e, bool eOdd) {
    float s1[4][2][2];
    stage1((const float*)(smem + (size_t)buf * 2048u), s1);
    float lo[4][2][2], hi[4][2][2];
    #pragma unroll
    for (int sb = 0; sb < 4; ++sb)
      #pragma unroll
      for (int p2 = 0; p2 < 2; ++p2)
        #pragma unroll
        for (int q2 = 0; q2 < 2; ++q2) {
          lo[sb][p2][q2] = (ev1[sb][p2][q2] + s1[sb][p2][q2]) * RC;
          hi[sb][p2][q2] = (ev1[sb][p2][q2] - s1[sb][p2][q2]) * RC;
        }
    // high1 = sub1[1:7] = {L·LH, L·HL, L·HH, H·LL, H·LH, H·HL, H·HH}
    {
      float* he = h1b + (size_t)e * 4096u;
      #pragma unroll
      for (int p2 = 0; p2 < 2; ++p2) {
        float* hp = he + (size_t)p2 * 64u;
#define H1ST(J, ARR) { v2f vv = { ARR[p2][0], ARR[p2][1] }; \
        __builtin_nontemporal_store(vv, (v2f*)(hp + (size_t)(J) * 4194304u)); }
        H1ST(0, lo[1]) H1ST(1, lo[2]) H1ST(2, lo[3])
        H1ST(3, hi[0]) H1ST(4, hi[1]) H1ST(5, hi[2]) H1ST(6, hi[3])
#undef H1ST
      }
    }
    // ---- level 2 (time index e) ----
    float s3[2][2];
    #pragma unroll
    for (int p2 = 0; p2 < 2; ++p2)
      #pragma unroll
      for (int q2 = 0; q2 < 2; ++q2)
        s3[p2][q2] = lif_step(vLLL[p2][q2], lo[0][p2][q2]);
    float sL2[2], sH2[2];
    #pragma unroll
    for (int q2 = 0; q2 < 2; ++q2) {
      sL2[q2] = lif_step(vL2[q2], (s3[0][q2] + s3[1][q2]) * RC);
      sH2[q2] = lif_step(vH2[q2], (s3[0][q2] - s3[1][q2]) * RC);
    }
    float s4[4];
    s4[0] = lif_step(vS2[0], (sL2[0] + sL2[1]) * RC);
    s4[1] = lif_step(vS2[1], (sL2[0] - sL2[1]) * RC);
    s4[2] = lif_step(vS2[2], (sH2[0] + sH2[1]) * RC);
    s4[3] = lif_step(vS2[3], (sH2[0] - sH2[1]) * RC);
    if (!eOdd) {
      #pragma unroll
      for (int k = 0; k < 4; ++k) ev2[k] = s4[k];
    } else {
      int f = e >> 1;
      float lo2[4], hi2[4];
      #pragma unroll
      for (int k = 0; k < 4; ++k) {
        lo2[k] = (ev2[k] + s4[k]) * RC;
        hi2[k] = (ev2[k] - s4[k]) * RC;
      }
      __builtin_nontemporal_store(lo2[0], o0b + (size_t)f * 1024u);
      float* hq = h2b + (size_t)f * 1024u;
      __builtin_nontemporal_store(lo2[1], hq + (size_t)0 * 524288u);
      __builtin_nontemporal_store(lo2[2], hq + (size_t)1 * 524288u);
      __builtin_nontemporal_store(lo2[3], hq + (size_t)2 * 524288u);
      __builtin_nontemporal_store(hi2[0], hq + (size_t)3 * 524288u);
      __builtin_nontemporal_store(hi2[1], hq + (size_t)4 * 524288u);
      __builtin_nontemporal_store(hi2[2], hq + (size_t)5 * 524288u);
      __builtin_nontemporal_store(hi2[3], hq + (size_t)6 * 524288u);
    }
  };

  // ---- prologue: prefetch d = 0..6 into buffers 0..6 (28 async ops/wave) ----
  uint64_t pf = 0;
  #pragma unroll
  for (int d0 = 0; d0 < 7; ++d0) { issue((uint32_t)d0, pf); pf += 65536ull; }

  // ---- steady state: step d = 8*oo + s, buffer = s&7, prefetch distance 7.
  // Single barrier per step: the issue at the end of step d rewrites buffer
  // (d+7)&7 == (d-1)&7, whose readers all passed this step's barrier already.
  for (int oo = 0; oo < 7; ++oo) {
    #pragma unroll
    for (int s = 0; s < 8; ++s) {
      WAITA(24);            // 7 tiles (28 ops) outstanding -> oldest tile done
      __syncthreads();      // tile d visible to all waves of the block
      if (s & 1) oddStep((uint32_t)s, 4 * oo + (s >> 1), ((s >> 1) & 1) != 0);
      else       evenStep((uint32_t)s);
      issue((uint32_t)((s + 7) & 7), pf); pf += 65536ull;   // prefetch d+7
    }
  }
  // ---- epilogue: d = 56..63, drain the pipeline ----
  WAITA(24); __syncthreads(); evenStep(0u);            issue(7u, pf); // d=63
  WAITA(24); __syncthreads(); oddStep(1u, 28, false);
  WAITA(20); __syncthreads(); evenStep(2u);
  WAITA(16); __syncthreads(); oddStep(3u, 29, true);
  WAITA(12); __syncthreads(); evenStep(4u);
  WAITA(8);  __syncthreads(); oddStep(5u, 30, false);
  WAITA(4);  __syncthreads(); evenStep(6u);
  WAITA(0);  __syncthreads(); oddStep(7u, 31, true);
}

extern "C" void kernel_launch(void* const* d_in, const int* in_sizes, int n_in,
                              void* d_out, int out_size, void* d_ws, size_t ws_size,
                              hipStream_t stream) {
  (void)in_sizes; (void)n_in; (void)out_size; (void)d_ws; (void)ws_size;
  const float* x = (const float*)d_in[0];
  float* out     = (float*)d_out;
  // 64 (n,c) slices x 8 P-strips = 512 blocks of 128 threads (4 wave32).
  dim3 grid(512), block(128);
  size_t shmem = 8u * 2048u * sizeof(float);   // 64 KB LDS ring (WGP has 320 KB)
  hipLaunchKernelGGL(dwt3d_lif_fused, grid, block, shmem, stream, x, out);
}